// SCAN_77292231459418
// MI455X (gfx1250) — compile-verified
//
#include <hip/hip_runtime.h>
#include <hip/hip_bf16.h>

typedef float v2f __attribute__((ext_vector_type(2)));
typedef float v8f __attribute__((ext_vector_type(8)));

#define NI 128   // images
#define NC 256   // captions
#define NR 36    // regions per image
#define NW 48    // words per caption
#define ND 256   // embed dim

#define SI_STRIDE 260   // padded row stride (floats) for 256-wide tiles
#define SS_STRIDE 52    // padded row stride (floats) for 48-wide tiles

#define OFF_I   0
#define OFF_C   (48 * SI_STRIDE)
#define OFF_S   (2 * 48 * SI_STRIDE)
#define OFF_T   (OFF_S + 48 * SS_STRIDE)
#define OFF_PP  (OFF_T + 48 * SS_STRIDE)   // [48][16] partial cap*wctx sums
#define OFF_PQ  (OFF_PP + 48 * 16)         // [48][16] partial wctx^2 sums
#define OFF_CSQ (OFF_PQ + 48 * 16)         // [48] caption sq norms
#define LDS_FLOATS (OFF_CSQ + 48)

#define WMMA_F32(A, B, C) \
  __builtin_amdgcn_wmma_f32_16x16x4_f32(false, (A), false, (B), (short)0, (C), false, false)

__global__ __launch_bounds__(128) void scan_xattn_kernel(
    const float* __restrict__ images,
    const float* __restrict__ captions,
    float* __restrict__ out)
{
  extern __shared__ float lds[];
  float* sI   = lds + OFF_I;    // [48][260]  image (rows 36..47 zero)
  float* sC   = lds + OFF_C;    // [48][260]  caption
  float* sS   = lds + OFF_S;    // [48][52]   attn scores, region-major
  float* sT   = lds + OFF_T;    // [48][52]   softmax weights, word-major
  float* pP   = lds + OFF_PP;   // [48][16]
  float* pQ   = lds + OFF_PQ;   // [48][16]
  float* sCsq = lds + OFF_CSQ;  // [48]

  const int tid  = threadIdx.x;
  const int lane = tid & 31;
  const int wv   = tid >> 5;      // wave id 0..3 (wave32)
  const int half = lane >> 4;     // K-half selector for WMMA operands
  const int l16  = lane & 15;
  const int i = blockIdx.x;       // image index
  const int c = blockIdx.y;       // caption index

  // ------- Phase A: async global->LDS staging (ASYNCcnt path, no VGPR hop) --
  {
    const char* gI = (const char*)(images   + (size_t)i * NR * ND);
    const char* gC = (const char*)(captions + (size_t)c * NW * ND);
    // image: 36 rows x 64 float4 = 2304 (18 per thread, exact)
    for (int idx = tid; idx < NR * (ND / 4); idx += 128) {
      int r = idx >> 6, q = idx & 63;
      unsigned lOff = (unsigned)(OFF_I * 4 + r * (SI_STRIDE * 4) + q * 16);
      unsigned gOff = (unsigned)idx * 16u;
      asm volatile("global_load_async_to_lds_b128 %0, %1, %2"
                   :: "v"(lOff), "v"(gOff), "s"(gI) : "memory");
    }
    // caption: 48 rows x 64 float4 = 3072 (24 per thread, exact)
    for (int idx = tid; idx < NW * (ND / 4); idx += 128) {
      int r = idx >> 6, q = idx & 63;
      unsigned lOff = (unsigned)(OFF_C * 4 + r * (SI_STRIDE * 4) + q * 16);
      unsigned gOff = (unsigned)idx * 16u;
      asm volatile("global_load_async_to_lds_b128 %0, %1, %2"
                   :: "v"(lOff), "v"(gOff), "s"(gC) : "memory");
    }
    // zero-pad image rows 36..47 (regular LDS stores)
    for (int idx = tid; idx < (48 - NR) * SI_STRIDE; idx += 128)
      sI[NR * SI_STRIDE + idx] = 0.0f;
    asm volatile("s_wait_asynccnt 0x0" ::: "memory");
  }
  __syncthreads();

  // ------- Phase B: S = I * C^T (48x48 padded, K=256), A cached in VGPRs ----
  // Waves 0..2 each own one tm row (3 tiles, same critical path as 3/2/2/2),
  // reusing the A fragments across all 3 tn tiles. Wave 3 does caption norms.
  if (wv < 3) {
    const int tm = wv;
    const float* aRow  = &sI[(tm * 16 + l16) * SI_STRIDE];
    const float* bRow0 = &sC[(0 * 16 + l16) * SI_STRIDE];
    const float* bRow1 = &sC[(1 * 16 + l16) * SI_STRIDE];
    const float* bRow2 = &sC[(2 * 16 + l16) * SI_STRIDE];
    v8f acc0 = {0,0,0,0,0,0,0,0}, acc1 = {0,0,0,0,0,0,0,0}, acc2 = {0,0,0,0,0,0,0,0};
    #pragma unroll 1
    for (int kh = 0; kh < ND; kh += 128) {
      v2f aF[32];
      #pragma unroll
      for (int kk = 0; kk < 32; ++kk) {
        int k0 = kh + kk * 4 + 2 * half;
        aF[kk].x = aRow[k0]; aF[kk].y = aRow[k0 + 1];
      }
      #pragma unroll
      for (int kk = 0; kk < 32; ++kk) {
        int k0 = kh + kk * 4 + 2 * half;
        v2f b; b.x = bRow0[k0]; b.y = bRow0[k0 + 1];
        acc0 = WMMA_F32(aF[kk], b, acc0);
      }
      #pragma unroll
      for (int kk = 0; kk < 32; ++kk) {
        int k0 = kh + kk * 4 + 2 * half;
        v2f b; b.x = bRow1[k0]; b.y = bRow1[k0 + 1];
        acc1 = WMMA_F32(aF[kk], b, acc1);
      }
      #pragma unroll
      for (int kk = 0; kk < 32; ++kk) {
        int k0 = kh + kk * 4 + 2 * half;
        v2f b; b.x = bRow2[k0]; b.y = bRow2[k0 + 1];
        acc2 = WMMA_F32(aF[kk], b, acc2);
      }
    }
    float* d0 = &sS[(tm * 16 + half * 8) * SS_STRIDE + l16];
    #pragma unroll
    for (int j = 0; j < 8; ++j) {
      d0[j * SS_STRIDE +  0] = acc0[j];
      d0[j * SS_STRIDE + 16] = acc1[j];
      d0[j * SS_STRIDE + 32] = acc2[j];
    }
  } else {
    // wave 3: caption squared norms, overlapped with GEMM1
    for (int w = tid - 96; w < NW; w += 32) {
      float s = 0.0f;
      const float* cw = &sC[w * SI_STRIDE];
      #pragma unroll 4
      for (int d = 0; d < ND; ++d) s += cw[d] * cw[d];
      sCsq[w] = s;
    }
  }
  __syncthreads();

  // ------- Phase C: leaky_relu -> l2norm over words -> softmax over regions -
  if (tid < NR) {
    float* row = &sS[tid * SS_STRIDE];
    float ss = 0.0f;
    #pragma unroll
    for (int w = 0; w < NW; ++w) {
      float x = row[w];
      x = x > 0.0f ? x : 0.1f * x;
      row[w] = x;
      ss += x * x;
    }
    float inv = 1.0f / (sqrtf(ss) + 1e-8f);
    #pragma unroll
    for (int w = 0; w < NW; ++w) row[w] *= inv;
  }
  __syncthreads();
  if (tid < NW) {   // per-word softmax over regions (|x|<=1 so exp(9x) safe)
    int w = tid;
    float s = 0.0f;
    #pragma unroll
    for (int r = 0; r < NR; ++r) {
      float e = __expf(9.0f * sS[r * SS_STRIDE + w]);
      sT[w * SS_STRIDE + r] = e;
      s += e;
    }
    float inv = 1.0f / s;
    #pragma unroll
    for (int r = 0; r < NR; ++r) sT[w * SS_STRIDE + r] *= inv;
    #pragma unroll
    for (int r = NR; r < 48; ++r) sT[w * SS_STRIDE + r] = 0.0f;
  }
  __syncthreads();

  // ------- Phase D: wctx = attn * I (48x256, K=48), A cached per tm ---------
  // Partial reductions written once per (word, d-tile) slot: deterministic,
  // no atomics, no init required.
  #pragma unroll 1
  for (int tm = 0; tm < 3; ++tm) {
    const float* aRow = &sT[(tm * 16 + l16) * SS_STRIDE];
    v2f aF[12];
    #pragma unroll
    for (int kk = 0; kk < 12; ++kk) {
      int k0 = kk * 4 + 2 * half;
      aF[kk].x = aRow[k0]; aF[kk].y = aRow[k0 + 1];
    }
    #pragma unroll 1
    for (int tn = wv; tn < 16; tn += 4) {
      const int dcol = tn * 16 + l16;
      v8f acc = {0,0,0,0,0,0,0,0};
      #pragma unroll
      for (int kk = 0; kk < 12; ++kk) {
        int k0 = kk * 4 + 2 * half;
        v2f b;
        b.x = sI[k0 * SI_STRIDE + dcol];
        b.y = sI[(k0 + 1) * SI_STRIDE + dcol];
        acc = WMMA_F32(aF[kk], b, acc);
      }
      const int wbase = tm * 16 + half * 8;
      #pragma unroll
      for (int j = 0; j < 8; ++j) {
        float val = acc[j];
        float cwv = sC[(wbase + j) * SI_STRIDE + dcol];
        float p = val * cwv;
        float q = val * val;
        #pragma unroll
        for (int m = 1; m <= 8; m <<= 1) {  // sum the 16 lanes sharing a word
          p += __shfl_xor(p, m, 32);
          q += __shfl_xor(q, m, 32);
        }
        if (l16 == 0) {
          pP[(wbase + j) * 16 + tn] = p;
          pQ[(wbase + j) * 16 + tn] = q;
        }
      }
    }
  }
  __syncthreads();

  // ------- Phase E: cosine sims (fixed-order partial sums) + LogSumExp ------
  if (tid < NW) {
    float w12 = 0.0f, w2 = 0.0f;
    #pragma unroll
    for (int tn = 0; tn < 16; ++tn) {
      w12 += pP[tid * 16 + tn];
      w2  += pQ[tid * 16 + tn];
    }
    float denom = fmaxf(sqrtf(sCsq[tid]) * sqrtf(w2), 1e-8f);
    sCsq[tid] = 6.0f * (w12 / denom);   // overwrite with scaled row_sim
  }
  __syncthreads();
  if (tid == 0) {
    float m = -3.4e38f;
    #pragma unroll 4
    for (int w = 0; w < NW; ++w) m = fmaxf(m, sCsq[w]);
    float s = 0.0f;
    #pragma unroll 4
    for (int w = 0; w < NW; ++w) s += __expf(sCsq[w] - m);
    out[(size_t)i * NC + c] = (m + __logf(s)) * (1.0f / 6.0f);
  }
}

extern "C" void kernel_launch(void* const* d_in, const int* in_sizes, int n_in,
                              void* d_out, int out_size, void* d_ws, size_t ws_size,
                              hipStream_t stream) {
  (void)in_sizes; (void)n_in; (void)out_size; (void)d_ws; (void)ws_size;
  const float* images   = (const float*)d_in[0];  // (128, 36, 256) f32
  const float* captions = (const float*)d_in[1];  // (256, 48, 256) f32
  float* out = (float*)d_out;                     // (128, 256) f32

  dim3 grid(NI, NC);
  dim3 block(128);
  size_t shmem = (size_t)LDS_FLOATS * sizeof(float);  // ~126 KB of 320 KB/WGP
  scan_xattn_kernel<<<grid, block, shmem, stream>>>(images, captions, out);
}